// PCENFrontend_90202903150895
// MI455X (gfx1250) — compile-verified
//
#include <hip/hip_runtime.h>
#include <hip/hip_bf16.h>
#include <math.h>

// ---------------- problem constants ----------------
#define T_IN       320000
#define BATCH      16
#define NFFT       1024
#define HOP        320
#define NBINS      513          // NFFT/2 + 1
#define KPAD       516          // NBINS padded to multiple of 4 for WMMA K
#define KSTEPS     (KPAD / 4)   // 129 WMMA K-steps
#define NMELS      128
#define NFRAMES    1001         // 1 + (320000 + 1024 - 1024)/320
#define FRAMES_PAD 1008         // NFRAMES padded to multiple of 16 for WMMA M
#define FTILES     63           // FRAMES_PAD / 16
#define PCEN_EPS   1e-6f
#define LN_EPS     1e-6f

typedef __attribute__((ext_vector_type(2))) float v2f;
typedef __attribute__((ext_vector_type(8))) float v8f;

// =====================================================================
// Kernel A: frame + reflect pad + Hann window + 1024-pt FFT (LDS) + power
// One 256-thread workgroup per frame. Grid covers padded frames; pad
// frames are zero-filled so the WMMA GEMM never reads poisoned memory.
// =====================================================================
__global__ void fft_power_kernel(const float* __restrict__ x,
                                 float* __restrict__ power) {
    const int fg  = blockIdx.x;                 // 0 .. BATCH*FRAMES_PAD-1
    const int b   = fg / FRAMES_PAD;
    const int f   = fg % FRAMES_PAD;
    const int tid = threadIdx.x;                // 0..255

    float* prow = power + ((size_t)b * FRAMES_PAD + f) * KPAD;

    if (f >= NFRAMES) {                         // zero pad frames
        for (int k = tid; k < KPAD; k += 256) prow[k] = 0.0f;
        return;
    }

    __shared__ float re[NFFT];
    __shared__ float im[NFFT];

    const float* xb = x + (size_t)b * T_IN;

    // load with reflect padding + Hann window, bit-reversed placement
    for (int j = tid; j < NFFT; j += 256) {
        int s = f * HOP + j - (NFFT / 2);
        if (s < 0)      s = -s;                        // numpy 'reflect'
        if (s >= T_IN)  s = 2 * (T_IN - 1) - s;
        float w  = 0.5f * (1.0f - __cosf(6.28318530717958647692f *
                                         (float)j * (1.0f / (float)NFFT)));
        unsigned rj = __brev((unsigned)j) >> (32 - 10);
        re[rj] = xb[s] * w;
        im[rj] = 0.0f;
    }
    __syncthreads();

    // iterative radix-2 DIT FFT, 10 stages, 512 butterflies/stage
    for (int st = 1; st <= 10; ++st) {
        const int m    = 1 << st;
        const int half = m >> 1;
        for (int t = tid; t < (NFFT / 2); t += 256) {
            int j   = t & (half - 1);
            int pos = ((t >> (st - 1)) << st) + j;
            float ang = -6.28318530717958647692f * (float)j / (float)m;
            float sn, cs;
            __sincosf(ang, &sn, &cs);
            float ur = re[pos],        ui = im[pos];
            float vr = re[pos + half], vi = im[pos + half];
            float tr = vr * cs - vi * sn;
            float ti = vr * sn + vi * cs;
            re[pos]        = ur + tr;  im[pos]        = ui + ti;
            re[pos + half] = ur - tr;  im[pos + half] = ui - ti;
        }
        __syncthreads();
    }

    // power spectrum, bins 0..512; zero the K-pad lanes 513..515
    for (int k = tid; k < KPAD; k += 256) {
        float p = 0.0f;
        if (k < NBINS) p = re[k] * re[k] + im[k] * im[k];
        prow[k] = p;
    }
}

// =====================================================================
// Kernel pack_fb: repack the mel filterbank (513 x 128, K-major) into
// WMMA B-fragment order so the GEMM's B loads are single b64 per frag:
//   fbPack[((ks*8 + t)*32 + lane)*2 + {0,1}] =
//       fb[4*ks + 2*(lane>>4) + {0,1}][t*16 + (lane&15)]   (zero past K=512)
// Size = 129*8*32*2 floats = 66,048 floats (264 KB, L2-resident).
// =====================================================================
__global__ void pack_fb_kernel(const float* __restrict__ fb,
                               float* __restrict__ fbPack) {
    int i = blockIdx.x * blockDim.x + threadIdx.x;   // 0 .. 33,023
    if (i < KSTEPS * 8 * 32) {
        int ks   = i >> 8;          // K-step (256 fragment-pairs per step)
        int rem  = i & 255;
        int t    = rem >> 5;        // N tile 0..7
        int lane = rem & 31;
        int half = lane >> 4;
        int n    = t * 16 + (lane & 15);
        int k    = 4 * ks + 2 * half;
        float v0 = (k     < NBINS) ? fb[(size_t)k * NMELS + n]       : 0.0f;
        float v1 = (k + 1 < NBINS) ? fb[(size_t)(k + 1) * NMELS + n] : 0.0f;
        fbPack[(size_t)i * 2]     = v0;
        fbPack[(size_t)i * 2 + 1] = v1;
    }
}

// =====================================================================
// Kernel B: mel projection GEMM via V_WMMA_F32_16X16X4_F32.
// One wave32 owns one (batch, 16-frame) tile and all 128 mel columns
// (8 accumulator tiles of 16x16 = 64 acc VGPRs). 129 K-steps; per step:
// 1x b64 A load + 8x b64 B loads (pre-packed) + 8 WMMA.
//
// A-frag layout (ISA 7.12.2, 32-bit A 16x4): lanes 0-15 -> {K0,K1},
// lanes 16-31 -> {K2,K3}, M = lane&15.
// C/D layout: lanes 0-15 -> M = vgpr, lanes 16-31 -> M = vgpr+8, N = lane&15.
// EXEC must be all-ones at the WMMA: no branching before the stores.
// =====================================================================
__global__ void mel_wmma_kernel(const float* __restrict__ power,
                                const float* __restrict__ fbPack,
                                float* __restrict__ E) {
    const int wave = (int)((blockIdx.x * blockDim.x + threadIdx.x) >> 5); // 0..1007
    const int lane = threadIdx.x & 31;
    const int b    = wave / FTILES;
    const int ft   = wave % FTILES;
    const int half = lane >> 4;        // 0: lanes 0-15, 1: lanes 16-31
    const int l15  = lane & 15;

    // A pointer: row = frame (M), base K offset folds in the lane-half K shift
    const float* Arow = power + ((size_t)b * FRAMES_PAD + (size_t)ft * 16 + l15) * KPAD
                              + (half << 1);
    // packed B pointer for this lane
    const float* Bp = fbPack + (size_t)lane * 2;

    v8f acc[8];
#pragma unroll
    for (int t = 0; t < 8; ++t) acc[t] = (v8f){0,0,0,0,0,0,0,0};

    for (int ks = 0; ks < KSTEPS; ++ks) {
        v2f a = *(const v2f*)(Arow + (size_t)ks * 4);      // 8B-aligned
        const float* Bk = Bp + (size_t)ks * (8 * 32 * 2);
#pragma unroll
        for (int t = 0; t < 8; ++t) {
            v2f bb = *(const v2f*)(Bk + t * 64);           // 8B-aligned b64
            acc[t] = __builtin_amdgcn_wmma_f32_16x16x4_f32(
                         /*neg_a=*/false, a, /*neg_b=*/false, bb,
                         /*c_mod=*/(short)0, acc[t],
                         /*reuse_a=*/false, /*reuse_b=*/false);
        }
    }

    // store D tiles (divergent frame-validity guard is fine after the WMMAs)
    float* Ebase = E + ((size_t)b * FRAMES_PAD + (size_t)ft * 16) * NMELS;
#pragma unroll
    for (int t = 0; t < 8; ++t) {
#pragma unroll
        for (int r = 0; r < 8; ++r) {
            int frRow = r + (half << 3);              // M within tile
            if (ft * 16 + frRow < NFRAMES)
                Ebase[(size_t)frRow * NMELS + t * 16 + l15] = acc[t][r];
        }
    }
}

// =====================================================================
// Kernel C1: PCEN temporal IIR smoother. 2048 independent (b, mel)
// channels; lanes within a wave cover consecutive mels -> coalesced
// 128-float row reads per step. M[0] = E[0] (scan carry init = E[0]).
// =====================================================================
__global__ void pcen_iir_kernel(const float* __restrict__ E,
                                float* __restrict__ S,
                                const float* __restrict__ log_s) {
    const int idx = blockIdx.x * blockDim.x + threadIdx.x;   // 0..2047
    const int b   = idx >> 7;
    const int m   = idx & 127;
    const float s = __expf(log_s[0]);
    const float om = 1.0f - s;

    const float* Eb = E + (size_t)b * FRAMES_PAD * NMELS + m;
    float*       Sb = S + (size_t)b * FRAMES_PAD * NMELS + m;

    float mm = Eb[0];
    Sb[0] = mm;
    for (int f = 1; f < NFRAMES; ++f) {
        mm = om * mm + s * Eb[(size_t)f * NMELS];
        Sb[(size_t)f * NMELS] = mm;
    }
}

// =====================================================================
// Kernel C2: PCEN nonlinearity + LayerNorm over the 128-mel axis.
// One 128-thread workgroup per (batch, frame) row; LDS tree reductions.
// =====================================================================
__global__ void pcen_ln_kernel(const float* __restrict__ E,
                               const float* __restrict__ S,
                               const float* __restrict__ log_alpha,
                               const float* __restrict__ log_delta,
                               const float* __restrict__ log_r,
                               float* __restrict__ out) {
    const int fg = blockIdx.x;               // 0 .. BATCH*NFRAMES-1
    const int b  = fg / NFRAMES;
    const int f  = fg % NFRAMES;
    const int m  = threadIdx.x;              // 0..127

    const float alpha = __expf(log_alpha[0]);
    const float delta = __expf(log_delta[0]);
    const float r     = __expf(log_r[0]);

    const size_t off = ((size_t)b * FRAMES_PAD + f) * NMELS + m;
    const float e  = E[off];
    const float sm = S[off];

    float denom = __powf(PCEN_EPS + sm, alpha);
    float p = __powf(e / denom + delta, r) - __powf(delta, r);

    __shared__ float red[NMELS];
    red[m] = p;
    __syncthreads();
    for (int st = 64; st > 0; st >>= 1) {
        if (m < st) red[m] += red[m + st];
        __syncthreads();
    }
    const float mu = red[0] * (1.0f / (float)NMELS);
    __syncthreads();

    const float d = p - mu;
    red[m] = d * d;
    __syncthreads();
    for (int st = 64; st > 0; st >>= 1) {
        if (m < st) red[m] += red[m + st];
        __syncthreads();
    }
    const float var = red[0] * (1.0f / (float)NMELS);

    out[((size_t)b * NFRAMES + f) * NMELS + m] = d * rsqrtf(var + LN_EPS);
}

// =====================================================================
// Launch. Workspace layout (floats):
//   fbPack : KSTEPS*8*32*2                    =    66,048
//   power  : BATCH*FRAMES_PAD*KPAD            = 8,322,048
//   E      : BATCH*FRAMES_PAD*NMELS           = 2,064,384
//   S      : BATCH*FRAMES_PAD*NMELS           = 2,064,384
// total ~ 50.1 MB
// =====================================================================
extern "C" void kernel_launch(void* const* d_in, const int* in_sizes, int n_in,
                              void* d_out, int out_size, void* d_ws, size_t ws_size,
                              hipStream_t stream) {
    const float* x         = (const float*)d_in[0];
    const float* fb        = (const float*)d_in[1];
    const float* log_s     = (const float*)d_in[2];
    const float* log_alpha = (const float*)d_in[3];
    const float* log_delta = (const float*)d_in[4];
    const float* log_r     = (const float*)d_in[5];
    float* out = (float*)d_out;

    float* fbPack = (float*)d_ws;
    float* power  = fbPack + (size_t)KSTEPS * 8 * 32 * 2;
    float* E      = power  + (size_t)BATCH * FRAMES_PAD * KPAD;
    float* S      = E      + (size_t)BATCH * FRAMES_PAD * NMELS;

    pack_fb_kernel<<<(KSTEPS * 8 * 32 + 255) / 256, 256, 0, stream>>>(fb, fbPack);

    fft_power_kernel<<<BATCH * FRAMES_PAD, 256, 0, stream>>>(x, power);

    // 16 batches * 63 frame tiles = 1008 waves = 126 blocks of 8 waves
    mel_wmma_kernel<<<(BATCH * FTILES) / 8, 256, 0, stream>>>(power, fbPack, E);

    pcen_iir_kernel<<<(BATCH * NMELS) / 256, 256, 0, stream>>>(E, S, log_s);

    pcen_ln_kernel<<<BATCH * NFRAMES, NMELS, 0, stream>>>(E, S, log_alpha,
                                                          log_delta, log_r, out);
}